// LocalLSHAttention_59167469470173
// MI455X (gfx1250) — compile-verified
//
#include <hip/hip_runtime.h>
#include <hip/hip_bf16.h>
#include <math.h>

typedef __attribute__((ext_vector_type(2))) float v2f;
typedef __attribute__((ext_vector_type(8))) float v8f;

#define B_ 2
#define S_ 2048
#define D_ 128
#define NH_ 8
#define HALF_NB_ 16
#define EPS_ 1e-5f
#define SCALE_ 0.08838834764831845f   // 1/sqrt(128)

// ---------------------------------------------------------------------------
// Kernel 1: layernorm + LSH bucket (last hash round only). One wave32 per row.
// Writes xm = layernorm(x)*mask and bucket ids to workspace.
// ---------------------------------------------------------------------------
__global__ __launch_bounds__(256) void k_ln_hash(
    const float* __restrict__ x, const float* __restrict__ mask,
    const float* __restrict__ rotations, float* __restrict__ xm,
    int* __restrict__ bkt)
{
  const int row  = (blockIdx.x * blockDim.x + threadIdx.x) >> 5;
  const int lane = threadIdx.x & 31;
  if (row >= B_ * S_) return;
  const int b = row / S_;

  const float* xr = x + (size_t)row * D_;
  float v[4];
#pragma unroll
  for (int i = 0; i < 4; ++i) v[i] = xr[lane + 32 * i];

  float s = v[0] + v[1] + v[2] + v[3];
#pragma unroll
  for (int off = 16; off >= 1; off >>= 1) s += __shfl_xor(s, off, 32);
  const float mu = s * (1.0f / D_);

  float q = 0.f;
#pragma unroll
  for (int i = 0; i < 4; ++i) { v[i] -= mu; q += v[i] * v[i]; }
#pragma unroll
  for (int off = 16; off >= 1; off >>= 1) q += __shfl_xor(q, off, 32);
  const float rstd = rsqrtf(q * (1.0f / D_) + EPS_);
  const float mk   = mask[row];

#pragma unroll
  for (int i = 0; i < 4; ++i) v[i] *= rstd;            // xn
#pragma unroll
  for (int i = 0; i < 4; ++i)
    xm[(size_t)row * D_ + lane + 32 * i] = v[i] * mk;  // xm = xn*mask

  // rot_k = sum_d xn_d * R[b, d, NH-1, k], k = 0..15 (only last hash round used)
  float rk[HALF_NB_];
#pragma unroll
  for (int k = 0; k < HALF_NB_; ++k) rk[k] = 0.f;
#pragma unroll
  for (int i = 0; i < 4; ++i) {
    const int d = lane + 32 * i;
    const float4* Rr = (const float4*)(rotations +
        ((((size_t)b * D_ + d) * NH_) + (NH_ - 1)) * HALF_NB_);
#pragma unroll
    for (int c = 0; c < 4; ++c) {
      float4 rv = Rr[c];
      rk[4*c+0] += v[i] * rv.x; rk[4*c+1] += v[i] * rv.y;
      rk[4*c+2] += v[i] * rv.z; rk[4*c+3] += v[i] * rv.w;
    }
  }
#pragma unroll
  for (int k = 0; k < HALF_NB_; ++k) {
#pragma unroll
    for (int off = 16; off >= 1; off >>= 1) rk[k] += __shfl_xor(rk[k], off, 32);
  }
  if (lane == 0) {
    // argmax over concat([rot, -rot]) with first-occurrence tie-breaking
    float best = rk[0]; int bi = 0;
#pragma unroll
    for (int k = 1; k < HALF_NB_; ++k) if (rk[k] > best) { best = rk[k]; bi = k; }
#pragma unroll
    for (int k = 0; k < HALF_NB_; ++k) if (-rk[k] > best) { best = -rk[k]; bi = HALF_NB_ + k; }
    bkt[row] = bi;
  }
}

// ---------------------------------------------------------------------------
// Kernel 2: bucket-masked attention via fp32 WMMA (16x16x4).
// One block (4 wave32) per 16-query tile; waves split the 128 key tiles.
// Non-matching columns contribute exp(0)=1 to denominator, 0 to numerator.
// ---------------------------------------------------------------------------
__global__ __launch_bounds__(128) void k_bucket_attn(
    const float* __restrict__ xm, const int* __restrict__ bkt,
    float* __restrict__ out)
{
  __shared__ float lds_p[4][16][17];     // per-wave P transpose buffer (padded)
  __shared__ float lds_num[16][D_];      // cross-wave numerator accumulator
  __shared__ float lds_den[16];          // cross-wave denominator accumulator

  const int b   = blockIdx.x >> 7;       // S_/16 = 128 query tiles per batch
  const int s0  = (blockIdx.x & 127) * 16;
  const int wid  = threadIdx.x >> 5;
  const int lane = threadIdx.x & 31;
  const int r    = lane & 15;            // row-in-tile / column-in-tile index
  const int hi   = lane >> 4;
  const int kk   = hi * 2;               // K sub-offset for 16x16x4 A/B fragments

  for (int i = threadIdx.x; i < 16 * D_; i += 128) (&lds_num[0][0])[i] = 0.f;
  if (threadIdx.x < 16) lds_den[threadIdx.x] = 0.f;
  __syncthreads();

  const size_t base = (size_t)b * S_ * D_;
  const float* Qrow = xm + base + (size_t)(s0 + r) * D_;

  // Q fragments in A layout for all 32 K-chunks: lane r holds {Q[r][4c+kk], Q[r][4c+kk+1]}
  v2f qa[32];
#pragma unroll
  for (int c = 0; c < 32; ++c)
    qa[c] = *(const v2f*)(Qrow + 4 * c + kk);

  // query bucket ids in C-layout rows: m = rr + hi*8
  int qb[8];
#pragma unroll
  for (int rr = 0; rr < 8; ++rr) qb[rr] = bkt[b * S_ + s0 + rr + hi * 8];

  v8f acc[8];                            // out numerator, 8 d-blocks of 16
#pragma unroll
  for (int nb = 0; nb < 8; ++nb) acc[nb] = v8f{0.f,0.f,0.f,0.f,0.f,0.f,0.f,0.f};
  v8f dacc = v8f{0.f,0.f,0.f,0.f,0.f,0.f,0.f,0.f};

  for (int j = wid; j < S_ / 16; j += 4) {
    const int t0 = j * 16;
    const float* Krow = xm + base + (size_t)(t0 + r) * D_;
    const int kb = bkt[b * S_ + t0 + r];           // bucket of key column n=r

    // scores: S = Q * K^T  (32 fp32 WMMAs, K=4 each)
    v8f cfrag = v8f{0.f,0.f,0.f,0.f,0.f,0.f,0.f,0.f};
#pragma unroll
    for (int c = 0; c < 32; ++c) {
      v2f kf = *(const v2f*)(Krow + 4 * c + kk);
      cfrag = __builtin_amdgcn_wmma_f32_16x16x4_f32(false, qa[c], false, kf,
                                                    (short)0, cfrag, false, false);
    }

    // masked exp: non-match -> (num 0, den 1); match -> (e, e)
    float pnum[8];
#pragma unroll
    for (int rr = 0; rr < 8; ++rr) {
      const float e = __expf(cfrag[rr] * SCALE_);
      const bool match = (qb[rr] == kb);
      pnum[rr]  = match ? e : 0.0f;
      dacc[rr] += match ? e : 1.0f;
    }

    // transpose P (C layout -> A layout) through per-wave LDS tile
    __syncthreads();
#pragma unroll
    for (int rr = 0; rr < 8; ++rr) lds_p[wid][rr + hi * 8][r] = pnum[rr];
    __syncthreads();

    v2f pa[4];
#pragma unroll
    for (int c2 = 0; c2 < 4; ++c2) {
      pa[c2].x = lds_p[wid][r][4 * c2 + kk];
      pa[c2].y = lds_p[wid][r][4 * c2 + kk + 1];
    }

    // out += P * V   (8 d-blocks x 4 K-chunks = 32 fp32 WMMAs)
#pragma unroll
    for (int nb = 0; nb < 8; ++nb) {
      v8f o = acc[nb];
#pragma unroll
      for (int c2 = 0; c2 < 4; ++c2) {
        const int row0 = t0 + 4 * c2 + kk;
        v2f vb;
        vb.x = xm[base + (size_t)row0       * D_ + nb * 16 + r];
        vb.y = xm[base + (size_t)(row0 + 1) * D_ + nb * 16 + r];
        o = __builtin_amdgcn_wmma_f32_16x16x4_f32(false, pa[c2], false, vb,
                                                  (short)0, o, false, false);
      }
      acc[nb] = o;
    }
  }

  // cross-wave reduction in LDS (ds_add_f32)
#pragma unroll
  for (int nb = 0; nb < 8; ++nb)
#pragma unroll
    for (int rr = 0; rr < 8; ++rr)
      atomicAdd(&lds_num[rr + hi * 8][nb * 16 + r], acc[nb][rr]);
#pragma unroll
  for (int rr = 0; rr < 8; ++rr)
    atomicAdd(&lds_den[rr + hi * 8], dacc[rr]);
  __syncthreads();

  // finalize: out = num / den
  for (int i = threadIdx.x; i < 16 * D_; i += 128) {
    const int m = i >> 7, d = i & 127;
    out[base + (size_t)(s0 + m) * D_ + d] = lds_num[m][d] / lds_den[m];
  }
}

// ---------------------------------------------------------------------------
extern "C" void kernel_launch(void* const* d_in, const int* in_sizes, int n_in,
                              void* d_out, int out_size, void* d_ws, size_t ws_size,
                              hipStream_t stream) {
  (void)in_sizes; (void)n_in; (void)out_size; (void)ws_size;
  const float* x    = (const float*)d_in[0];
  const float* mask = (const float*)d_in[1];
  const float* rot  = (const float*)d_in[2];
  // d_in[3] = num_buckets scalar (structure is fixed: 32 buckets, 16 rotations)
  float* out = (float*)d_out;

  float* xm  = (float*)d_ws;                                          // 2 MB
  int*   bkt = (int*)((char*)d_ws + (size_t)B_ * S_ * D_ * sizeof(float)); // 16 KB

  k_ln_hash<<<(B_ * S_) / 8, 256, 0, stream>>>(x, mask, rot, xm, bkt);
  k_bucket_attn<<<B_ * (S_ / 16), 128, 0, stream>>>(xm, bkt, out);
}